// TFGNN_64639257805189
// MI455X (gfx1250) — compile-verified
//
#include <hip/hip_runtime.h>
#include <hip/hip_bf16.h>

// ---------------------------------------------------------------------------
// TFGNN forward for MI455X (gfx1250).
//   1. coeff_kernel        : Taylor coeffs from alpha/beta (fp64 on device)
//   2. memset deg/cnt
//   3. deg_count_kernel    : deg[src] (non-loop), in-degree histogram by dst
//   4. dinv_kernel         : deg -> rsqrt(deg) in place
//   5. scan_kernel         : exclusive scan -> row_ptr, fill_pos (1 block)
//   6. csr_fill_kernel     : pack (src, w=-dinv[s]*dinv[d]) as int2 CSR by dst
//   7. mlp_wmma_kernel     : relu(feat@W1+b1)@W2+b2 via v_wmma_f32_16x16x4_f32
//                            -> TxA = x, hidden = (1+coeff0)*x
//   8. hop_kernel x10      : Tx' = gather-sum(w*Tx[src]) per dst (CSR, no
//                            atomics, float2 lanes), hidden += coeff[d]*Tx'.
//                            diag==0, s==1 exactly (w_off<=0 -> max(w_off,1)=1)
//   9. logsoftmax_kernel   : row-wise log_softmax over 64 classes (float2)
// All hot buffers (~103 MB) are L2-resident on the 192 MB L2.
// ---------------------------------------------------------------------------

typedef __attribute__((ext_vector_type(2))) float v2f;
typedef __attribute__((ext_vector_type(8))) float v8f;

#define F_IN 500
#define HID  64
#define NCLS 64
#define KFRQ 5
#define DORD 10

// ---------------------------------------------------------------------------
// 1. coeff[d] = sum_k alpha[k]*SINC[k][d] + beta[k]*COSC[k][d]
// ---------------------------------------------------------------------------
__global__ void tfgnn_coeff_kernel(const float* __restrict__ alpha,
                                   const float* __restrict__ beta,
                                   float* __restrict__ coeff) {
    int d = threadIdx.x;
    if (d > DORD) return;
    double acc = 0.0;
    for (int k = 0; k < KFRQ; ++k) {
        double xk = (double)(k + 1) * 3.14159265358979323846; // OMEGA=1
        double p = 1.0;                       // xk^d / d!
        for (int i = 1; i <= d; ++i) p *= xk / (double)i;
        double sinc = 0.0, cosc = 0.0;
        if ((d & 1) == 0) cosc = (((d / 2) & 1) ? -1.0 : 1.0) * p;
        else              sinc = ((((d - 1) / 2) & 1) ? -1.0 : 1.0) * p;
        acc += (double)alpha[k] * sinc + (double)beta[k] * cosc;
    }
    coeff[d] = (float)acc;
}

// ---------------------------------------------------------------------------
// 3. Degree (by src, non-loop) and in-degree histogram (by dst, non-loop).
// ---------------------------------------------------------------------------
__global__ void tfgnn_deg_count_kernel(const int* __restrict__ src,
                                       const int* __restrict__ dst,
                                       float* __restrict__ deg,
                                       int* __restrict__ cnt, int e) {
    int i = blockIdx.x * blockDim.x + threadIdx.x;
    if (i >= e) return;
    int s = src[i], d = dst[i];
    if (s != d) {
        atomicAdd(&deg[s], 1.0f);
        atomicAdd(&cnt[d], 1);
    }
}

// 4. deg -> dinv in place
__global__ void tfgnn_dinv_kernel(float* __restrict__ deg, int n) {
    int i = blockIdx.x * blockDim.x + threadIdx.x;
    if (i >= n) return;
    float g = deg[i];
    deg[i] = (g > 0.0f) ? __frsqrt_rn(g) : 0.0f;
}

// ---------------------------------------------------------------------------
// 5. One-block exclusive scan of cnt[n] -> row_ptr[n+1], fill_pos[n].
// ---------------------------------------------------------------------------
__global__ __launch_bounds__(256)
void tfgnn_scan_kernel(const int* __restrict__ cnt, int* __restrict__ row_ptr,
                       int* __restrict__ fill_pos, int n) {
    __shared__ int sums[256];
    int t = threadIdx.x;
    int chunk = (n + 255) / 256;
    int start = t * chunk;
    int end = start + chunk; if (end > n) end = n;
    int s = 0;
    for (int i = start; i < end; ++i) s += cnt[i];
    sums[t] = s;
    __syncthreads();
    if (t == 0) {
        int acc = 0;
        for (int i = 0; i < 256; ++i) { int v = sums[i]; sums[i] = acc; acc += v; }
        row_ptr[n] = acc;
    }
    __syncthreads();
    int acc = sums[t];
    for (int i = start; i < end; ++i) {
        row_ptr[i] = acc;
        fill_pos[i] = acc;
        acc += cnt[i];
    }
}

// ---------------------------------------------------------------------------
// 6. Scatter edges into dst-indexed packed CSR: entry = {src, bits(w)} so the
//    hop kernel does a single b64 load per edge.
// ---------------------------------------------------------------------------
__global__ void tfgnn_csr_fill_kernel(const int* __restrict__ src,
                                      const int* __restrict__ dst,
                                      const float* __restrict__ dinv,
                                      int* __restrict__ fill_pos,
                                      int2* __restrict__ csr, int e) {
    int i = blockIdx.x * blockDim.x + threadIdx.x;
    if (i >= e) return;
    int s = src[i], d = dst[i];
    if (s != d) {
        float w = -dinv[s] * dinv[d];      // s_scale == 1 exactly (w_off <= 0)
        int pos = atomicAdd(&fill_pos[d], 1);
        csr[pos] = make_int2(s, __float_as_int(w));
    }
}

// ---------------------------------------------------------------------------
// 7. Fused MLP with fp32 WMMA. One wave owns a 16-row node tile x 64 cols
//    (4 N-tiles of 16). 8 waves / block -> 128 rows per block.
//    Row indices are clamped (not masked) so EXEC is all-ones at every WMMA.
// ---------------------------------------------------------------------------
__global__ __launch_bounds__(256)
void tfgnn_mlp_wmma_kernel(const float* __restrict__ feat,
                           const float* __restrict__ W1,
                           const float* __restrict__ b1,
                           const float* __restrict__ W2,
                           const float* __restrict__ b2,
                           const float* __restrict__ coeff,
                           float* __restrict__ x_out,
                           float* __restrict__ hidden, int n) {
    __shared__ float lh[8][16 * HID];               // 32 KB: per-wave h tiles
    const int wave = threadIdx.x >> 5;
    const int lane = threadIdx.x & 31;
    const int mrow = lane & 15;                     // M index within tile
    const int khalf = (lane < 16) ? 0 : 2;          // K sub-offset per A/B frag
    const int rbase = (blockIdx.x * 8 + wave) * 16; // first node row of tile

    int arow = rbase + mrow;                        // clamped load row
    if (arow >= n) arow = n - 1;

    // ---- GEMM1: acc[t] = feat_tile @ W1[:, t*16..t*16+15] --------------
    v8f acc[4];
#pragma unroll
    for (int t = 0; t < 4; ++t) acc[t] = (v8f)0.0f;

    const float* aptr = feat + (size_t)arow * F_IN;
    for (int k0 = 0; k0 < F_IN; k0 += 4) {
        const int kk = k0 + khalf;
        v2f a; a.x = aptr[kk]; a.y = aptr[kk + 1];
#pragma unroll
        for (int t = 0; t < 4; ++t) {
            const int col = t * 16 + mrow;
            v2f b;
            b.x = W1[(size_t)kk * HID + col];
            b.y = W1[(size_t)(kk + 1) * HID + col];
            acc[t] = __builtin_amdgcn_wmma_f32_16x16x4_f32(
                false, a, false, b, (short)0, acc[t], false, false);
        }
    }

    // ---- bias + ReLU, stash h tile in LDS ------------------------------
    float* myh = &lh[wave][0];
#pragma unroll
    for (int t = 0; t < 4; ++t) {
        const int col = t * 16 + mrow;
        const float bb = b1[col];
#pragma unroll
        for (int j = 0; j < 8; ++j) {
            const int r = j + ((lane < 16) ? 0 : 8);
            float v = acc[t][j] + bb;
            myh[r * HID + col] = (v > 0.0f) ? v : 0.0f;
        }
    }
    __syncthreads();

    // ---- GEMM2: acc2[t] = h_tile @ W2[:, t*16..t*16+15] ----------------
    v8f acc2[4];
#pragma unroll
    for (int t = 0; t < 4; ++t) acc2[t] = (v8f)0.0f;

    for (int k0 = 0; k0 < HID; k0 += 4) {
        const int kk = k0 + khalf;
        v2f a; a.x = myh[mrow * HID + kk]; a.y = myh[mrow * HID + kk + 1];
#pragma unroll
        for (int t = 0; t < 4; ++t) {
            const int col = t * 16 + mrow;
            v2f b;
            b.x = W2[kk * NCLS + col];
            b.y = W2[(kk + 1) * NCLS + col];
            acc2[t] = __builtin_amdgcn_wmma_f32_16x16x4_f32(
                false, a, false, b, (short)0, acc2[t], false, false);
        }
    }

    // ---- epilogue: x = acc2 + b2; TxA = x; hidden = (1+coeff0)*x -------
    const float c0p1 = 1.0f + coeff[0];
#pragma unroll
    for (int t = 0; t < 4; ++t) {
        const int col = t * 16 + mrow;
        const float bb = b2[col];
#pragma unroll
        for (int j = 0; j < 8; ++j) {
            const int r = j + ((lane < 16) ? 0 : 8);
            const int row = rbase + r;
            if (row < n) {
                float v = acc2[t][j] + bb;
                size_t o = (size_t)row * NCLS + col;
                x_out[o] = v;
                hidden[o] = c0p1 * v;
            }
        }
    }
}

// ---------------------------------------------------------------------------
// 8. One diffusion hop: per-node CSR gather (no atomics). One wave per node,
//    lane owns channel pair {2*lane, 2*lane+1}: one b64 load per edge for the
//    packed CSR entry + one b64 load of the Tx[src] row slice; b64 stores.
//    hidden += coeff[d] * Tx' fused.
// ---------------------------------------------------------------------------
__global__ __launch_bounds__(256)
void tfgnn_hop_kernel(const float* __restrict__ txp, float* __restrict__ txn,
                      float* __restrict__ hidden,
                      const int* __restrict__ row_ptr,
                      const int2* __restrict__ csr,
                      const float* __restrict__ coeff, int d, int n) {
    const int node = (int)((blockIdx.x * blockDim.x + threadIdx.x) >> 5);
    const int lane = threadIdx.x & 31;
    if (node >= n) return;
    const int beg = row_ptr[node], end = row_ptr[node + 1];
    const float c = coeff[d];
    float a0 = 0.0f, a1 = 0.0f;
    for (int i = beg; i < end; ++i) {
        const int2 ew = csr[i];                       // {src, bits(w)}
        const float w = __int_as_float(ew.y);
        const float2 v =
            *(const float2*)(txp + (size_t)ew.x * NCLS + 2 * lane);
        a0 += w * v.x;
        a1 += w * v.y;
    }
    const size_t o = (size_t)node * NCLS + 2 * lane;
    *(float2*)(txn + o) = make_float2(a0, a1);
    float2 h = *(const float2*)(hidden + o);
    h.x += c * a0;
    h.y += c * a1;
    *(float2*)(hidden + o) = h;
}

// ---------------------------------------------------------------------------
// 9. Row-wise log_softmax over 64 classes; one wave per node, float2 lanes.
// ---------------------------------------------------------------------------
__global__ __launch_bounds__(256)
void tfgnn_logsoftmax_kernel(const float* __restrict__ hidden,
                             float* __restrict__ out, int n) {
    const int node = (int)((blockIdx.x * blockDim.x + threadIdx.x) >> 5);
    const int lane = threadIdx.x & 31;
    if (node >= n) return;
    const size_t o = (size_t)node * NCLS + 2 * lane;
    const float2 v = *(const float2*)(hidden + o);
    float m = fmaxf(v.x, v.y);
    for (int k = 16; k > 0; k >>= 1) m = fmaxf(m, __shfl_xor(m, k, 32));
    float e = expf(v.x - m) + expf(v.y - m);
    for (int k = 16; k > 0; k >>= 1) e += __shfl_xor(e, k, 32);
    const float l = logf(e);
    *(float2*)(out + o) = make_float2(v.x - m - l, v.y - m - l);
}

// ---------------------------------------------------------------------------
// Host-side launch. Inputs (setup_inputs order):
//   0 feature [N,500] f32   1 edges [2,E] int   2 W1 [500,64] f32
//   3 b1 [64] f32           4 W2 [64,64] f32    5 b2 [64] f32
//   6 alpha [1,5] f32       7 beta [1,5] f32
// Output: [N,64] f32 log-probs.
// ---------------------------------------------------------------------------
extern "C" void kernel_launch(void* const* d_in, const int* in_sizes, int n_in,
                              void* d_out, int out_size, void* d_ws, size_t ws_size,
                              hipStream_t stream) {
    const float* feat  = (const float*)d_in[0];
    const int*   edges = (const int*)d_in[1];
    const float* W1    = (const float*)d_in[2];
    const float* b1    = (const float*)d_in[3];
    const float* W2    = (const float*)d_in[4];
    const float* b2    = (const float*)d_in[5];
    const float* alpha = (const float*)d_in[6];
    const float* beta  = (const float*)d_in[7];
    float* out = (float*)d_out;

    const int N = in_sizes[0] / F_IN;
    const int E = in_sizes[1] / 2;
    const int* e_src = edges;
    const int* e_dst = edges + E;

    // ---- workspace carve-up (256 B aligned) ----
    char* ws = (char*)d_ws;
    size_t off = 0;
    auto carve = [&](size_t bytes) -> char* {
        char* p = ws + off;
        off = (off + bytes + 255) & ~(size_t)255;
        return p;
    };
    float* coeff    = (float*)carve((DORD + 1) * sizeof(float));
    float* deg      = (float*)carve((size_t)N * sizeof(float));   // -> dinv
    int*   cnt      = (int*)  carve((size_t)N * sizeof(int));
    int*   row_ptr  = (int*)  carve((size_t)(N + 1) * sizeof(int));
    int*   fill_pos = (int*)  carve((size_t)N * sizeof(int));
    int2*  csr      = (int2*) carve((size_t)E * sizeof(int2));
    float* txA      = (float*)carve((size_t)N * NCLS * sizeof(float));
    float* txB      = (float*)carve((size_t)N * NCLS * sizeof(float));
    float* hidden   = (float*)carve((size_t)N * NCLS * sizeof(float));
    (void)ws_size; (void)n_in; (void)out_size;

    const int eb = (E + 255) / 256;
    const int nb = (N + 255) / 256;
    const int waveb = (N + 7) / 8;                    // 8 waves (nodes) / block
    const int tiles = (N + 15) / 16;
    const int mlpb = (tiles + 7) / 8;                 // 8 row-tiles / block

    // 1. coefficients
    tfgnn_coeff_kernel<<<1, 32, 0, stream>>>(alpha, beta, coeff);
    // 2. zero deg + cnt
    hipMemsetAsync(deg, 0, (size_t)N * sizeof(float), stream);
    hipMemsetAsync(cnt, 0, (size_t)N * sizeof(int), stream);
    // 3-6. graph preprocessing -> packed CSR by dst
    tfgnn_deg_count_kernel<<<eb, 256, 0, stream>>>(e_src, e_dst, deg, cnt, E);
    tfgnn_dinv_kernel<<<nb, 256, 0, stream>>>(deg, N);
    tfgnn_scan_kernel<<<1, 256, 0, stream>>>(cnt, row_ptr, fill_pos, N);
    tfgnn_csr_fill_kernel<<<eb, 256, 0, stream>>>(e_src, e_dst, deg, fill_pos,
                                                  csr, E);
    // 7. MLP (WMMA): txA = x, hidden = (1+coeff0)*x
    tfgnn_mlp_wmma_kernel<<<mlpb, 256, 0, stream>>>(feat, W1, b1, W2, b2, coeff,
                                                    txA, hidden, N);
    // 8. ten diffusion hops, ping-pong txA/txB (diag term is exactly 0)
    float* tp = txA; float* tn = txB;
    for (int d = 1; d <= DORD; ++d) {
        tfgnn_hop_kernel<<<waveb, 256, 0, stream>>>(tp, tn, hidden, row_ptr,
                                                    csr, coeff, d, N);
        float* t = tp; tp = tn; tn = t;
    }
    // 9. log_softmax -> out
    tfgnn_logsoftmax_kernel<<<waveb, 256, 0, stream>>>(hidden, out, N);
}